// Unet_82016695484898
// MI455X (gfx1250) — compile-verified
//
#include <hip/hip_runtime.h>
#include <math.h>

// ---------------------------------------------------------------------------
// VMamba SS2D block for MI455X (gfx1250, wave32).
//   GEMMs      -> V_WMMA_F32_16X16X4_F32 (full-precision WMMA)
//   Scan       -> 96 wave32 blocks, fwd+bwd direction fused per lane
//   Everything stays in the 192MB L2 (~147MB workspace).
// ---------------------------------------------------------------------------

typedef __attribute__((ext_vector_type(2))) float v2f;
typedef __attribute__((ext_vector_type(8))) float v8f;

#define D_MODEL 96
#define D_INNER 192
#define D_STATE 16
#define DT_RANK 6
#define BB      8
#define HH      64
#define WW      64
#define LL      4096
#define XDBL_LD 40   // 38 (=6+16+16) padded to 40

// scan position t of direction k  ->  canonical spatial index l = h*64+w
__device__ __forceinline__ int canon_pos(int k, int t) {
  int p = (k >= 2) ? (LL - 1 - t) : t;
  if (k & 1) p = ((p & 63) << 6) | (p >> 6);   // 64x64 transpose = index swap
  return p;
}

// ---------------------------------------------------------------------------
// Kernel 1: in_proj  xz[b,l,e] = sum_c x[b,l,c] * W[e,c]   (M=32768,N=384,K=96)
// One wave computes one 16x16 tile of C via 24 chained f32 WMMAs.
// A 16x4 frag: lane m = lane&15, reg r holds K = r + 2*(lane>>4)
// B 4x16 frag: lane n = lane&15, reg r holds K = r + 2*(lane>>4)  (row-striped)
// The xc/z split is decided on block-uniform col0 (16-aligned boundary) so the
// store epilogue is divergence-free.
// ---------------------------------------------------------------------------
__global__ __launch_bounds__(32)
void k_in_proj(const float* __restrict__ x, const float* __restrict__ w,
               float* __restrict__ xc, float* __restrict__ z) {
  const int row0 = blockIdx.x * 16;          // over B*L
  const int col0 = blockIdx.y * 16;          // over 2*D_INNER
  const int lane = threadIdx.x;
  const int mn = lane & 15, hi = lane >> 4;
  const float* arow = x + (size_t)(row0 + mn) * D_MODEL;
  const float* brow = w + (size_t)(col0 + mn) * D_MODEL;
  v8f acc = {};
#pragma unroll 8
  for (int k0 = 0; k0 < D_MODEL; k0 += 4) {
    v2f a = { arow[k0 + 2 * hi], arow[k0 + 2 * hi + 1] };
    v2f b = { brow[k0 + 2 * hi], brow[k0 + 2 * hi + 1] };
    acc = __builtin_amdgcn_wmma_f32_16x16x4_f32(false, a, false, b,
                                                (short)0, acc, false, false);
  }
  // Uniform select of destination buffer + column base (no lane divergence).
  float* dst = (col0 < D_INNER) ? xc : z;
  const int cbase = (col0 < D_INNER) ? col0 : (col0 - D_INNER);
#pragma unroll
  for (int r = 0; r < 8; ++r)
    dst[(size_t)(row0 + r + 8 * hi) * D_INNER + cbase + mn] = acc[r];
}

// ---------------------------------------------------------------------------
// Kernel 2: depthwise 3x3 conv (SAME, zero pad) + bias + SiLU.
// Block = one (b,h,w) pixel, 192 threads = channels (coalesced over d).
// ---------------------------------------------------------------------------
__global__ __launch_bounds__(192)
void k_conv_silu(const float* __restrict__ xc, const float* __restrict__ cw,
                 const float* __restrict__ cb, float* __restrict__ xa) {
  const int bl = blockIdx.x;                 // b*L + l
  const int d  = threadIdx.x;
  const int l  = bl & (LL - 1);
  const int b  = bl >> 12;
  const int h  = l >> 6, w = l & 63;
  float wv[9];
#pragma unroll
  for (int t = 0; t < 9; ++t) wv[t] = cw[d * 9 + t];
  float acc = cb[d];
#pragma unroll
  for (int i = -1; i <= 1; ++i) {
    const int hh = h + i;
    if (hh < 0 || hh >= HH) continue;
#pragma unroll
    for (int j = -1; j <= 1; ++j) {
      const int ww2 = w + j;
      if (ww2 < 0 || ww2 >= WW) continue;
      acc = fmaf(wv[(i + 1) * 3 + (j + 1)],
                 xc[(size_t)((b << 12) + (hh << 6) + ww2) * D_INNER + d], acc);
    }
  }
  const float s = 1.0f / (1.0f + __expf(-acc));
  xa[(size_t)bl * D_INNER + d] = acc * s;
}

// ---------------------------------------------------------------------------
// Kernel 3: x_dbl[b,k,t,c] = sum_d xa[b, canon(k,t), d] * xw[k,c,d]
// (M=32768 per k, N=38 padded to 48, K=192). xs is never materialized —
// the A rows are gathered through canon_pos.
// ---------------------------------------------------------------------------
__global__ __launch_bounds__(32)
void k_xdbl(const float* __restrict__ xa, const float* __restrict__ xw,
            float* __restrict__ xdbl) {
  const int t0   = (blockIdx.x & 255) * 16;  // scan positions within one b
  const int b    = blockIdx.x >> 8;
  const int col0 = blockIdx.y * 16;          // 0,16,32 (N padded to 48)
  const int k    = blockIdx.z;
  const int lane = threadIdx.x;
  const int mn = lane & 15, hi = lane >> 4;
  const float* arow = xa + (size_t)((b << 12) + canon_pos(k, t0 + mn)) * D_INNER;
  const int  c      = col0 + mn;
  const bool bval   = (c < 38);
  const float bm    = bval ? 1.0f : 0.0f;
  const float* brow = xw + (size_t)(k * 38 + (bval ? c : 0)) * D_INNER;
  v8f acc = {};
#pragma unroll 8
  for (int k0 = 0; k0 < D_INNER; k0 += 4) {
    v2f a  = { arow[k0 + 2 * hi], arow[k0 + 2 * hi + 1] };
    v2f bb = { bm * brow[k0 + 2 * hi], bm * brow[k0 + 2 * hi + 1] };
    acc = __builtin_amdgcn_wmma_f32_16x16x4_f32(false, a, false, bb,
                                                (short)0, acc, false, false);
  }
#pragma unroll
  for (int r = 0; r < 8; ++r) {
    const int t  = t0 + r + 8 * hi;
    const int cc = col0 + mn;
    if (cc < 38)
      xdbl[((size_t)(b * 4 + k) * LL + t) * XDBL_LD + cc] = acc[r];
  }
}

// ---------------------------------------------------------------------------
// Kernel 4: selective scan. One wave32 block = (b, direction pair, 32-ch chunk).
// Each lane runs forward (k=pair) and backward (k=pair+2) scans together;
// dt_proj + softplus folded in (6 MACs). Writes go to the pair's canonical
// y buffer at disjoint positions -> no atomics, deterministic.
// ---------------------------------------------------------------------------
__global__ __launch_bounds__(32)
void k_scan(const float* __restrict__ xa, const float* __restrict__ xdbl,
            const float* __restrict__ A_logs, const float* __restrict__ Ds,
            const float* __restrict__ dtw, const float* __restrict__ dtb,
            float* __restrict__ y0, float* __restrict__ y1) {
  const int d    = blockIdx.x * 32 + threadIdx.x;   // 0..191
  const int pair = blockIdx.y;                      // 0: k={0,2}, 1: k={1,3}
  const int b    = blockIdx.z;
  const int kf = pair, kb = pair + 2;
  float* yb = pair ? y1 : y0;

  float Af[16], Ab[16], hf[16], hb[16];
#pragma unroll
  for (int n = 0; n < 16; ++n) {
    Af[n] = -__expf(A_logs[(size_t)(kf * D_INNER + d) * 16 + n]);
    Ab[n] = -__expf(A_logs[(size_t)(kb * D_INNER + d) * 16 + n]);
    hf[n] = 0.0f; hb[n] = 0.0f;
  }
  const float Dfv = Ds[kf * D_INNER + d], Dbv = Ds[kb * D_INNER + d];
  float wtf[6], wtb[6];
#pragma unroll
  for (int r = 0; r < 6; ++r) {
    wtf[r] = dtw[(size_t)(kf * D_INNER + d) * 6 + r];
    wtb[r] = dtw[(size_t)(kb * D_INNER + d) * 6 + r];
  }
  const float btf = dtb[kf * D_INNER + d], btb = dtb[kb * D_INNER + d];

  const float* vf_base = xdbl + (size_t)(b * 4 + kf) * LL * XDBL_LD;
  const float* vb_base = xdbl + (size_t)(b * 4 + kb) * LL * XDBL_LD;

  for (int t = 0; t < LL; ++t) {
    const int Pf = canon_pos(kf, t);
    const int Pb = canon_pos(kb, t);
    const float* vf = vf_base + (size_t)t * XDBL_LD;   // wave-uniform rows
    const float* vb = vb_base + (size_t)t * XDBL_LD;
    const float uf = xa[(size_t)((b << 12) + Pf) * D_INNER + d];
    const float ub = xa[(size_t)((b << 12) + Pb) * D_INNER + d];
    if (t + 16 < LL) {                                  // global_prefetch_b8
      __builtin_prefetch(&xa[(size_t)((b << 12) + canon_pos(kf, t + 16)) * D_INNER + d], 0, 0);
      __builtin_prefetch(&xa[(size_t)((b << 12) + canon_pos(kb, t + 16)) * D_INNER + d], 0, 0);
      __builtin_prefetch(vf + 16 * XDBL_LD, 0, 0);
      __builtin_prefetch(vb + 16 * XDBL_LD, 0, 0);
    }
    float df = btf, db = btb;
#pragma unroll
    for (int r = 0; r < 6; ++r) {
      df = fmaf(wtf[r], vf[r], df);
      db = fmaf(wtb[r], vb[r], db);
    }
    df = (df > 20.0f) ? df : log1pf(__expf(df));        // softplus
    db = (db > 20.0f) ? db : log1pf(__expf(db));
    const float duf = df * uf, dub = db * ub;
    float yfv = 0.0f, ybv = 0.0f;
#pragma unroll
    for (int n = 0; n < 16; ++n) {
      const float dAf = __expf(df * Af[n]);
      const float dAb = __expf(db * Ab[n]);
      hf[n] = fmaf(dAf, hf[n], duf * vf[6 + n]);
      hb[n] = fmaf(dAb, hb[n], dub * vb[6 + n]);
      yfv = fmaf(hf[n], vf[22 + n], yfv);
      ybv = fmaf(hb[n], vb[22 + n], ybv);
    }
    yfv = fmaf(uf, Dfv, yfv);
    ybv = fmaf(ub, Dbv, ybv);
    yb[(size_t)((b << 12) + Pf) * D_INNER + d] = yfv;
    yb[(size_t)((b << 12) + Pb) * D_INNER + d] = ybv;
  }
}

// ---------------------------------------------------------------------------
// Kernel 5: y = y0+y1; LayerNorm over D_INNER; multiply by SiLU(z). In-place
// into y0 (each thread only rewrites its own element).
// ---------------------------------------------------------------------------
__global__ __launch_bounds__(192)
void k_ln_gate(float* __restrict__ y0, const float* __restrict__ y1,
               const float* __restrict__ z, const float* __restrict__ g,
               const float* __restrict__ bt) {
  __shared__ float s1[192], s2[192];
  const int bl = blockIdx.x;
  const int d  = threadIdx.x;
  const float v = y0[(size_t)bl * D_INNER + d] + y1[(size_t)bl * D_INNER + d];
  s1[d] = v; s2[d] = v * v;
  __syncthreads();
  if (d < 64) {
    s1[d] += s1[d + 64] + s1[d + 128];
    s2[d] += s2[d + 64] + s2[d + 128];
  }
  __syncthreads();
  for (int off = 32; off > 0; off >>= 1) {
    if (d < off) { s1[d] += s1[d + off]; s2[d] += s2[d + off]; }
    __syncthreads();
  }
  const float mu  = s1[0] * (1.0f / 192.0f);
  const float var = s2[0] * (1.0f / 192.0f) - mu * mu;
  const float inv = rsqrtf(var + 1e-5f);
  const float zn  = z[(size_t)bl * D_INNER + d];
  const float sig = 1.0f / (1.0f + __expf(-zn));
  y0[(size_t)bl * D_INNER + d] = ((v - mu) * inv * g[d] + bt[d]) * (zn * sig);
}

// ---------------------------------------------------------------------------
// Kernel 6: out_proj  out[b,l,c] = sum_d yg[b,l,d]*W[c,d]  (M=32768,N=96,K=192)
// ---------------------------------------------------------------------------
__global__ __launch_bounds__(32)
void k_out_proj(const float* __restrict__ yg, const float* __restrict__ w,
                float* __restrict__ out) {
  const int row0 = blockIdx.x * 16;
  const int col0 = blockIdx.y * 16;
  const int lane = threadIdx.x;
  const int mn = lane & 15, hi = lane >> 4;
  const float* arow = yg + (size_t)(row0 + mn) * D_INNER;
  const float* brow = w  + (size_t)(col0 + mn) * D_INNER;
  v8f acc = {};
#pragma unroll 8
  for (int k0 = 0; k0 < D_INNER; k0 += 4) {
    v2f a = { arow[k0 + 2 * hi], arow[k0 + 2 * hi + 1] };
    v2f b = { brow[k0 + 2 * hi], brow[k0 + 2 * hi + 1] };
    acc = __builtin_amdgcn_wmma_f32_16x16x4_f32(false, a, false, b,
                                                (short)0, acc, false, false);
  }
#pragma unroll
  for (int r = 0; r < 8; ++r)
    out[(size_t)(row0 + r + 8 * hi) * D_MODEL + col0 + mn] = acc[r];
}

// ---------------------------------------------------------------------------
extern "C" void kernel_launch(void* const* d_in, const int* in_sizes, int n_in,
                              void* d_out, int out_size, void* d_ws, size_t ws_size,
                              hipStream_t stream) {
  const float* x         = (const float*)d_in[0];
  const float* in_proj_w = (const float*)d_in[1];
  const float* conv_w    = (const float*)d_in[2];
  const float* conv_b    = (const float*)d_in[3];
  const float* x_proj_w  = (const float*)d_in[4];
  const float* dt_proj_w = (const float*)d_in[5];
  const float* dt_proj_b = (const float*)d_in[6];
  const float* A_logs    = (const float*)d_in[7];
  const float* Ds        = (const float*)d_in[8];
  const float* ln_g      = (const float*)d_in[9];
  const float* ln_b      = (const float*)d_in[10];
  const float* out_w     = (const float*)d_in[11];

  float* ws = (float*)d_ws;
  const size_t SZ_BLD = (size_t)BB * LL * D_INNER;       // 6,291,456 floats
  float* xc   = ws;                                      // 25.2 MB
  float* z    = xc   + SZ_BLD;                           // 25.2 MB
  float* xa   = z    + SZ_BLD;                           // 25.2 MB
  float* y0   = xa   + SZ_BLD;                           // 25.2 MB
  float* y1   = y0   + SZ_BLD;                           // 25.2 MB
  float* xdbl = y1   + SZ_BLD;                           // 21.0 MB

  const int MT = (BB * LL) / 16;                         // 2048 row tiles

  k_in_proj  <<<dim3(MT, (2 * D_INNER) / 16), 32, 0, stream>>>(x, in_proj_w, xc, z);
  k_conv_silu<<<BB * LL, 192, 0, stream>>>(xc, conv_w, conv_b, xa);
  k_xdbl     <<<dim3(MT, 3, 4), 32, 0, stream>>>(xa, x_proj_w, xdbl);
  k_scan     <<<dim3(D_INNER / 32, 2, BB), 32, 0, stream>>>(xa, xdbl, A_logs, Ds,
                                                            dt_proj_w, dt_proj_b, y0, y1);
  k_ln_gate  <<<BB * LL, 192, 0, stream>>>(y0, y1, z, ln_g, ln_b);
  k_out_proj <<<dim3(MT, D_MODEL / 16), 32, 0, stream>>>(y0, out_w, (float*)d_out);
}